// EMMaskGenerator_48833778156246
// MI455X (gfx1250) — compile-verified
//
#include <hip/hip_runtime.h>
#include <math.h>

// ---------------------------------------------------------------------------
// EM mask generator for MI455X (gfx1250, wave32). B=32, C=3, H=W=256, K=4.
//
// Per EM step (5x):
//   stats_kernel : per-pixel E-step of the previous iteration recomputed in
//     registers (no gm traffic), softmax -> probs, then sufficient statistics
//     accumulated on the matrix pipe with V_WMMA_F32_16X16X4_F32:
//       acc1[16x16]: rows (k,a)=probs[k]*padded[a], cols {padded(4), sh(3)}
//                    -> ops2[k][a][b] (cols 0-3), ops1[k][c][a] (cols 4-6)
//       acc2[16x16]: rows (k,a)=probs[k]*sh[a] (12 live rows), cols 4-6
//                    -> Mss[k][a][c] = sum probs*sh_a*sh_c
//   solve_kernel : 128 threads, one per (u,k); 4x4 inverse, W = ops1n inv(ops2n),
//     shifted_sigma = (Mss - W ops1^T - ops1 W^T + W ops2 W^T)/S + eps I,
//     mus/sigmas from ops2 moments; 3x3 inverses + logdets.
// Finally final_kernel writes generated_masks (last E-step) and
// generated_images = sum_k exp(pm_k) * (W_k * padded).
// ---------------------------------------------------------------------------

#define B_IMG   32
#define K_COMP  4
#define RGB     3
#define HW      65536
#define EPSV    1.1920929e-07f
#define NSTAT_K 37                   // ops2(16) + ops1(12) + mss(9)
#define NSTAT_U (NSTAT_K * K_COMP)   // 148
#define NSTAT   (NSTAT_U * B_IMG)    // 4736
#define NPAR_K  36                   // W(12) invSS(9) ldSS mu(3) invS(9) ldS pad
#define NPAR_U  (NPAR_K * K_COMP)    // 144
#define NPAR    (NPAR_U * B_IMG)     // 4608

typedef float v2f __attribute__((ext_vector_type(2)));
typedef float v8f __attribute__((ext_vector_type(8)));

// Staging stride 68: operand-read bank = (4m + 4kc + 2h) & 63 -> each of the
// 32 b64 reads covers a distinct bank pair (all 64 banks once). Even -> 8B
// aligned ds_load_b64 / ds_store_b64.
#define LSTR 68

__global__ void zero_kernel(float* p, int n) {
    int i = blockIdx.x * 256 + threadIdx.x;
    if (i < n) p[i] = 0.f;
}

// E-step log-prob for one (pixel, k); Pk = 36-float param block.
__device__ inline float estep_lp(const float* __restrict__ Pk,
                                 float i0, float i1, float i2,
                                 float s0, float s1, float s2,
                                 float& m0, float& m1, float& m2)
{
    m0 = Pk[0]*i0 + Pk[1]*i1 + Pk[2]*i2  + Pk[3];
    m1 = Pk[4]*i0 + Pk[5]*i1 + Pk[6]*i2  + Pk[7];
    m2 = Pk[8]*i0 + Pk[9]*i1 + Pk[10]*i2 + Pk[11];
    const float d0 = s0 - m0, d1 = s1 - m1, d2 = s2 - m2;
    const float q1 = Pk[12]*d0*d0 + Pk[16]*d1*d1 + Pk[20]*d2*d2
                   + 2.f*(Pk[13]*d0*d1 + Pk[14]*d0*d2 + Pk[17]*d1*d2);
    const float e0 = i0 - Pk[22], e1 = i1 - Pk[23], e2 = i2 - Pk[24];
    const float q2 = Pk[25]*e0*e0 + Pk[29]*e1*e1 + Pk[33]*e2*e2
                   + 2.f*(Pk[26]*e0*e1 + Pk[27]*e0*e2 + Pk[30]*e1*e2);
    return -0.5f*(q1 + Pk[21]) - 0.5f*(q2 + Pk[34]);
}

// ---------------------------------------------------------------------------
// Stats: grid = (32 blocks/img, 32 imgs), 128 threads (4 waves).
// Wave owns 512 pixels = 8 iterations of 64 px (2 px/lane, b64 loads).
// FIRST=1: probs = softmax(pm). FIRST=0: probs = softmax(pm + estep(params)).
// ---------------------------------------------------------------------------
template<int FIRST>
__global__ void __launch_bounds__(128)
stats_kernel(const float* __restrict__ img, const float* __restrict__ shf,
             const float* __restrict__ pm,  const float* __restrict__ params,
             float* __restrict__ gstats)
{
    __shared__ float lA1[4][16][LSTR];   // rows (k,a): probs[k]*padded[a]
    __shared__ float lA2[4][16][LSTR];   // rows (k,a): probs[k]*sh[a]; 12..15 = 0
    __shared__ float lB [4][8][LSTR];    // cols: {i0,i1,i2,1,s0,s1,s2}
    __shared__ float sstats[NSTAT_U];
    __shared__ float P[K_COMP][NPAR_K];

    const int u    = blockIdx.y;
    const int tid  = threadIdx.x;
    const int wave = tid >> 5;
    const int lane = tid & 31;
    const int m    = lane & 15;      // WMMA A-row / B,C column index
    const int h    = lane >> 4;      // lane half

    for (int i = tid; i < NSTAT_U; i += 128) sstats[i] = 0.f;
    if (!FIRST)
        for (int i = tid; i < NPAR_U; i += 128)
            ((float*)P)[i] = params[u * NPAR_U + i];
    // Dead A2 rows -> genuine zeros (branchless operand reads later).
    for (int c2 = lane; c2 < LSTR; c2 += 32) {
        lA2[wave][12][c2] = 0.f; lA2[wave][13][c2] = 0.f;
        lA2[wave][14][c2] = 0.f; lA2[wave][15][c2] = 0.f;
    }
    __syncthreads();

    const size_t b3 = (size_t)u * RGB    * HW;
    const size_t b4 = (size_t)u * K_COMP * HW;

    v8f acc1 = {0.f,0.f,0.f,0.f,0.f,0.f,0.f,0.f};
    v8f acc2 = {0.f,0.f,0.f,0.f,0.f,0.f,0.f,0.f};

    int p = blockIdx.x * 2048 + wave * 512 + 2 * lane;   // 2 px / lane
    const int f = (m < 7) ? m : 6;                       // cols >= 7: don't-care

    #pragma unroll 1
    for (int it = 0; it < 8; ++it, p += 64) {
        v2f I[3], S[3], L[K_COMP];
        #pragma unroll
        for (int c = 0; c < 3; ++c) {
            I[c] = *(const v2f*)&img[b3 + c*HW + p];
            S[c] = *(const v2f*)&shf[b3 + c*HW + p];
        }
        #pragma unroll
        for (int k = 0; k < K_COMP; ++k)
            L[k] = *(const v2f*)&pm[b4 + k*HW + p];

        if (!FIRST) {   // add previous-step generated masks, recomputed here
            #pragma unroll
            for (int j = 0; j < 2; ++j) {
                float t0, t1, t2;
                #pragma unroll
                for (int k = 0; k < K_COMP; ++k)
                    L[k][j] += estep_lp(&P[k][0], I[0][j], I[1][j], I[2][j],
                                        S[0][j], S[1][j], S[2][j], t0, t1, t2);
            }
        }

        // softmax over k (+ EPS), per sub-pixel
        v2f pr[K_COMP];
        #pragma unroll
        for (int j = 0; j < 2; ++j) {
            const float mx = fmaxf(fmaxf(L[0][j], L[1][j]), fmaxf(L[2][j], L[3][j]));
            float e[K_COMP], sum = 0.f;
            #pragma unroll
            for (int k = 0; k < K_COMP; ++k) { e[k] = __expf(L[k][j] - mx); sum += e[k]; }
            const float rs = 1.f / sum;
            #pragma unroll
            for (int k = 0; k < K_COMP; ++k) pr[k][j] = e[k] * rs + EPSV;
        }

        // Stage operand columns (column index = pixel-in-iteration = 2*lane+j).
        #pragma unroll
        for (int k = 0; k < K_COMP; ++k) {
            #pragma unroll
            for (int a = 0; a < 3; ++a) {
                v2f va; va.x = pr[k].x * I[a].x; va.y = pr[k].y * I[a].y;
                *(v2f*)&lA1[wave][k*4 + a][2*lane] = va;
                v2f vs; vs.x = pr[k].x * S[a].x; vs.y = pr[k].y * S[a].y;
                *(v2f*)&lA2[wave][k*3 + a][2*lane] = vs;
            }
            *(v2f*)&lA1[wave][k*4 + 3][2*lane] = pr[k];
        }
        #pragma unroll
        for (int a = 0; a < 3; ++a) {
            *(v2f*)&lB[wave][a][2*lane]     = I[a];
            *(v2f*)&lB[wave][4 + a][2*lane] = S[a];
        }
        { v2f one; one.x = 1.f; one.y = 1.f; *(v2f*)&lB[wave][3][2*lane] = one; }

        // Wave-local sync: staging is per-wave; LDS ops of one wave are
        // processed in order, so waiting on our own DScnt suffices.
        asm volatile("s_wait_dscnt 0x0" ::: "memory");

        // Consume 64 pixels as 16 K-chunks of 4; 2 WMMAs per chunk.
        #pragma unroll
        for (int kc = 0; kc < 16; ++kc) {
            const int c = kc * 4 + 2 * h;          // K = vgpr + 2*laneHalf
            v2f a1 = *(const v2f*)&lA1[wave][m][c];
            v2f a2 = *(const v2f*)&lA2[wave][m][c];
            v2f bb = *(const v2f*)&lB [wave][f][c];
            acc1 = __builtin_amdgcn_wmma_f32_16x16x4_f32(
                       false, a1, false, bb, (short)0, acc1, false, false);
            acc2 = __builtin_amdgcn_wmma_f32_16x16x4_f32(
                       false, a2, false, bb, (short)0, acc2, false, false);
        }
        asm volatile("" ::: "memory");   // keep next stores after these loads
    }

    // Scatter accumulators: vgpr r holds M = r + 8*h, N = lane&15.
    const int N = m;
    if (N < 7) {
        #pragma unroll
        for (int r = 0; r < 8; ++r) {
            const int M = r + 8 * h;
            const int k = M >> 2, a = M & 3;
            int idx;
            if (N < 4) idx = k * NSTAT_K + a * 4 + N;             // ops2[k][a][N]
            else       idx = k * NSTAT_K + 16 + (N - 4) * 4 + a;  // ops1[k][N-4][a]
            atomicAdd(&sstats[idx], acc1[r]);
            if (M < 12 && N >= 4) {
                const int k2 = M / 3, a2 = M % 3;                 // Mss[k2][a2][N-4]
                atomicAdd(&sstats[k2 * NSTAT_K + 28 + a2 * 3 + (N - 4)], acc2[r]);
            }
        }
    }
    __syncthreads();
    for (int i = tid; i < NSTAT_U; i += 128)
        atomicAdd(&gstats[u * NSTAT_U + i], sstats[i]);
}

// ---------------------------------------------------------------------------
// Small linear algebra (straight-line; matrices are SPD + eps*I).
// ---------------------------------------------------------------------------
__device__ inline void inv4(const float M[4][4], float R[4][4]) {
    float A[4][8];
    #pragma unroll
    for (int r = 0; r < 4; ++r)
        #pragma unroll
        for (int c = 0; c < 4; ++c) { A[r][c] = M[r][c]; A[r][c+4] = (r == c) ? 1.f : 0.f; }
    #pragma unroll
    for (int i = 0; i < 4; ++i) {
        const float piv = 1.f / A[i][i];
        #pragma unroll
        for (int j = 0; j < 8; ++j) A[i][j] *= piv;
        #pragma unroll
        for (int r = 0; r < 4; ++r) {
            if (r == i) continue;
            const float fac = A[r][i];
            #pragma unroll
            for (int j = 0; j < 8; ++j) A[r][j] -= fac * A[i][j];
        }
    }
    #pragma unroll
    for (int r = 0; r < 4; ++r)
        #pragma unroll
        for (int c = 0; c < 4; ++c) R[r][c] = A[r][c+4];
}

__device__ inline float inv3_logdet(const float S[3][3], float R[3][3]) {
    const float c00 = S[1][1]*S[2][2] - S[1][2]*S[2][1];
    const float c01 = S[1][2]*S[2][0] - S[1][0]*S[2][2];
    const float c02 = S[1][0]*S[2][1] - S[1][1]*S[2][0];
    const float det = S[0][0]*c00 + S[0][1]*c01 + S[0][2]*c02;
    const float id  = 1.f / det;
    R[0][0] = c00 * id;
    R[0][1] = (S[0][2]*S[2][1] - S[0][1]*S[2][2]) * id;
    R[0][2] = (S[0][1]*S[1][2] - S[0][2]*S[1][1]) * id;
    R[1][0] = c01 * id;
    R[1][1] = (S[0][0]*S[2][2] - S[0][2]*S[2][0]) * id;
    R[1][2] = (S[0][2]*S[1][0] - S[0][0]*S[1][2]) * id;
    R[2][0] = c02 * id;
    R[2][1] = (S[0][1]*S[2][0] - S[0][0]*S[2][1]) * id;
    R[2][2] = (S[0][0]*S[1][1] - S[0][1]*S[1][0]) * id;
    return logf(fabsf(det));
}

// One thread per (u,k): 128 threads total.
__global__ void solve_kernel(const float* __restrict__ gstats, float* __restrict__ params) {
    const int t = threadIdx.x;
    if (t >= B_IMG * K_COMP) return;
    const int u = t >> 2, k = t & 3;
    const float* st = gstats + u * NSTAT_U + k * NSTAT_K;

    float ops2[4][4], ops1[3][4], mss[3][3];
    #pragma unroll
    for (int a = 0; a < 4; ++a)
        #pragma unroll
        for (int b = 0; b < 4; ++b) ops2[a][b] = st[a*4 + b];
    #pragma unroll
    for (int a = 0; a < 3; ++a)
        #pragma unroll
        for (int b = 0; b < 4; ++b) ops1[a][b] = st[16 + a*4 + b];
    #pragma unroll
    for (int a = 0; a < 3; ++a)
        #pragma unroll
        for (int b = 0; b < 3; ++b) mss[a][b] = st[28 + a*3 + b];

    const float rn = 1.f / (ops2[0][0] + ops2[1][1] + ops2[2][2] + ops2[3][3]);
    float o2n[4][4];
    #pragma unroll
    for (int a = 0; a < 4; ++a)
        #pragma unroll
        for (int b = 0; b < 4; ++b) o2n[a][b] = ops2[a][b] * rn + ((a == b) ? EPSV : 0.f);
    float inv[4][4];
    inv4(o2n, inv);

    float W[3][4];
    #pragma unroll
    for (int a = 0; a < 3; ++a)
        #pragma unroll
        for (int b = 0; b < 4; ++b) {
            float acc = 0.f;
            #pragma unroll
            for (int c = 0; c < 4; ++c) acc += (ops1[a][c] * rn) * inv[c][b];
            W[a][b] = acc;
        }

    const float rS = 1.f / ops2[3][3];   // 1 / sum(probs)

    float T[3][4];
    #pragma unroll
    for (int a = 0; a < 3; ++a)
        #pragma unroll
        for (int d = 0; d < 4; ++d) {
            float acc = 0.f;
            #pragma unroll
            for (int c = 0; c < 4; ++c) acc += W[a][c] * ops2[c][d];
            T[a][d] = acc;
        }
    float SS[3][3];
    #pragma unroll
    for (int a = 0; a < 3; ++a)
        #pragma unroll
        for (int b = 0; b < 3; ++b) {
            float acc = mss[a][b];
            #pragma unroll
            for (int d = 0; d < 4; ++d)
                acc += -W[b][d] * ops1[a][d] - W[a][d] * ops1[b][d] + T[a][d] * W[b][d];
            SS[a][b] = acc * rS + ((a == b) ? EPSV : 0.f);
        }
    float iSS[3][3];
    const float ldSS = inv3_logdet(SS, iSS);

    const float mu[3] = { ops2[0][3]*rS, ops2[1][3]*rS, ops2[2][3]*rS };
    float SG[3][3];
    #pragma unroll
    for (int a = 0; a < 3; ++a)
        #pragma unroll
        for (int b = 0; b < 3; ++b)
            SG[a][b] = ops2[a][b]*rS - mu[a]*mu[b] + ((a == b) ? EPSV : 0.f);
    float iSG[3][3];
    const float ldS = inv3_logdet(SG, iSG);

    float* P = params + (u * K_COMP + k) * NPAR_K;
    #pragma unroll
    for (int a = 0; a < 3; ++a)
        #pragma unroll
        for (int b = 0; b < 4; ++b) P[a*4 + b] = W[a][b];
    #pragma unroll
    for (int a = 0; a < 3; ++a)
        #pragma unroll
        for (int b = 0; b < 3; ++b) P[12 + a*3 + b] = iSS[a][b];
    P[21] = ldSS;
    P[22] = mu[0]; P[23] = mu[1]; P[24] = mu[2];
    #pragma unroll
    for (int a = 0; a < 3; ++a)
        #pragma unroll
        for (int b = 0; b < 3; ++b) P[25 + a*3 + b] = iSG[a][b];
    P[34] = ldS;
}

// ---------------------------------------------------------------------------
// Final E-step: writes generated_masks and generated_images.
// grid = (128 blocks/img, 32 imgs), 256 threads, 2 px/thread.
// ---------------------------------------------------------------------------
__global__ void __launch_bounds__(256)
final_kernel(const float* __restrict__ img, const float* __restrict__ shf,
             const float* __restrict__ pm,  const float* __restrict__ params,
             float* __restrict__ gm_out, float* __restrict__ gi_out)
{
    __shared__ float P[K_COMP][NPAR_K];
    const int u = blockIdx.y, tid = threadIdx.x;
    if (tid < NPAR_U) ((float*)P)[tid] = params[u * NPAR_U + tid];
    __syncthreads();

    const int p = (blockIdx.x * 256 + tid) * 2;
    const size_t b3 = (size_t)u * RGB    * HW;
    const size_t b4 = (size_t)u * K_COMP * HW;

    v2f I[3], S[3];
    #pragma unroll
    for (int c = 0; c < 3; ++c) {
        I[c] = *(const v2f*)&img[b3 + c*HW + p];
        S[c] = *(const v2f*)&shf[b3 + c*HW + p];
    }

    v2f gi[3];
    #pragma unroll
    for (int c = 0; c < 3; ++c) { gi[c].x = 0.f; gi[c].y = 0.f; }

    #pragma unroll
    for (int k = 0; k < K_COMP; ++k) {
        const v2f pmk = *(const v2f*)&pm[b4 + k*HW + p];
        v2f lp;
        #pragma unroll
        for (int j = 0; j < 2; ++j) {
            float m0, m1, m2;
            lp[j] = estep_lp(&P[k][0], I[0][j], I[1][j], I[2][j],
                             S[0][j], S[1][j], S[2][j], m0, m1, m2);
            const float w = __expf(pmk[j]);
            gi[0][j] += w * m0; gi[1][j] += w * m1; gi[2][j] += w * m2;
        }
        *(v2f*)&gm_out[b4 + k*HW + p] = lp;
    }
    #pragma unroll
    for (int c = 0; c < 3; ++c)
        *(v2f*)&gi_out[b3 + c*HW + p] = gi[c];
}

// ---------------------------------------------------------------------------
extern "C" void kernel_launch(void* const* d_in, const int* in_sizes, int n_in,
                              void* d_out, int out_size, void* d_ws, size_t ws_size,
                              hipStream_t stream) {
    const float* images  = (const float*)d_in[0];   // (B,3,H,W)
    const float* shifted = (const float*)d_in[1];   // (B,3,H,W)
    const float* pmasks  = (const float*)d_in[2];   // (B,4,H,W)

    float* gm = (float*)d_out;                                  // generated_masks
    float* gi = (float*)d_out + (size_t)B_IMG * K_COMP * HW;    // generated_images

    float* gstats = (float*)d_ws;          // 4736 floats
    float* params = gstats + NSTAT;        // 4608 floats

    for (int it = 0; it < 5; ++it) {
        zero_kernel<<<(NSTAT + 255) / 256, 256, 0, stream>>>(gstats, NSTAT);
        if (it == 0)
            stats_kernel<1><<<dim3(32, B_IMG), 128, 0, stream>>>(
                images, shifted, pmasks, params, gstats);
        else
            stats_kernel<0><<<dim3(32, B_IMG), 128, 0, stream>>>(
                images, shifted, pmasks, params, gstats);
        solve_kernel<<<1, 128, 0, stream>>>(gstats, params);
    }
    final_kernel<<<dim3(128, B_IMG), 256, 0, stream>>>(
        images, shifted, pmasks, params, gm, gi);
}